// ProposalTarget_19516331393682
// MI455X (gfx1250) — compile-verified
//
#include <hip/hip_runtime.h>
#include <hip/hip_bf16.h>
#include <math.h>

// ---------------- problem constants (match reference) ----------------
#define NUM_CLASSES_C   21
#define TOTAL_SAMPLES_C 512
#define MAX_POS_C       128
#define N_GT_MAX        256
#define NBUCKET         2048
#define CAND_CAP        4096

// output layout in d_out (floats, concatenated in return order)
#define ROIS_OFF 0
#define LBL_OFF  (TOTAL_SAMPLES_C * 4)                       // 2048
#define TGT_OFF  (LBL_OFF + TOTAL_SAMPLES_C)                 // 2560
#define INS_OFF  (TGT_OFF + TOTAL_SAMPLES_C * NUM_CLASSES_C * 4)  // 45568
#define OUTW_OFF (INS_OFF + TOTAL_SAMPLES_C * NUM_CLASSES_C * 4)  // 88576
#define OUT_TOTAL (OUTW_OFF + TOTAL_SAMPLES_C * NUM_CLASSES_C * 4) // 131584

// ---------------- CDNA5 async global->LDS staging --------------------
#if defined(__gfx1250__) && __has_builtin(__builtin_amdgcn_global_load_async_to_lds_b128)
#define HAVE_ASYNC_LDS 1
typedef int v4i_t __attribute__((vector_size(16)));
#else
#define HAVE_ASYNC_LDS 0
#endif

// ---------------- threefry2x32 replica of jax.random.uniform ---------
__device__ __forceinline__ unsigned rotl32(unsigned v, unsigned d) {
  return (v << d) | (v >> (32u - d));
}

// element i of jax.random.uniform(jax.random.key(42), (n,)), n even
__device__ float jax_priority(unsigned i, unsigned n) {
  const unsigned ks0 = 0u, ks1 = 42u;
  const unsigned ks2 = ks0 ^ ks1 ^ 0x1BD11BDAu;
  const unsigned H = n >> 1;
  const bool first = (i < H);
  unsigned x0 = first ? i : (i - H);
  unsigned x1 = x0 + H;
  x0 += ks0; x1 += ks1;
#define TF_R(r) { x0 += x1; x1 = rotl32(x1, r); x1 ^= x0; }
  TF_R(13) TF_R(15) TF_R(26) TF_R(6)
  x0 += ks1; x1 += ks2 + 1u;
  TF_R(17) TF_R(29) TF_R(16) TF_R(24)
  x0 += ks2; x1 += ks0 + 2u;
  TF_R(13) TF_R(15) TF_R(26) TF_R(6)
  x0 += ks0; x1 += ks1 + 3u;
  TF_R(17) TF_R(29) TF_R(16) TF_R(24)
  x0 += ks1; x1 += ks2 + 4u;
  TF_R(13) TF_R(15) TF_R(26) TF_R(6)
  x0 += ks2; x1 += ks0 + 5u;
#undef TF_R
  unsigned bits = first ? x0 : x1;
  return __uint_as_float((bits >> 9) | 0x3f800000u) - 1.0f;
}

__device__ __forceinline__ int bucket_of(float p) {
  int b = (int)(p * (float)NBUCKET);
  return b < 0 ? 0 : (b > NBUCKET - 1 ? NBUCKET - 1 : b);
}

// ---------------- kernel 0: zero scratch control region --------------
__global__ void k_init(int* __restrict__ p, int n) {
  int i = blockIdx.x * blockDim.x + threadIdx.x;
  if (i < n) p[i] = 0;
}

// ---------------- kernel 1: IoU max/argmax + priority + histogram ----
#define BLK 256
#define RPT 2

__global__ __launch_bounds__(BLK) void k_iou(
    const float4* __restrict__ rois, const float4* __restrict__ gt,
    int n, int m,
    float* __restrict__ maxiou, int* __restrict__ argm, float* __restrict__ prio,
    int* __restrict__ fgHist, int* __restrict__ bgHist)
{
  __shared__ float4 sgt[N_GT_MAX];
  __shared__ float  sAreaB[N_GT_MAX];
  const int t = threadIdx.x;
  const int base = blockIdx.x * (BLK * RPT) + t;
  const int mm = m < N_GT_MAX ? m : N_GT_MAX;

  // CDNA5: prefetch this thread's roi tile into the cache hierarchy
  {
    int pi = base < n ? base : (n - 1);
    __builtin_prefetch((const void*)&rois[pi], 0, 3);
  }

  // stage GT boxes into LDS (async copy path on gfx1250)
  if (t < mm) {
#if HAVE_ASYNC_LDS
    __builtin_amdgcn_global_load_async_to_lds_b128(
        (__attribute__((address_space(1))) v4i_t*)(float4*)&gt[t],
        (__attribute__((address_space(3))) v4i_t*)&sgt[t],
        0, 0);
#else
    sgt[t] = gt[t];
#endif
  }
#if HAVE_ASYNC_LDS
  asm volatile("s_wait_asynccnt 0" ::: "memory");
#endif
  __syncthreads();
  if (t < mm) {
    float4 b = sgt[t];
    sAreaB[t] = (b.z - b.x + 1.0f) * (b.w - b.y + 1.0f);
  }
  __syncthreads();

  float4 a[RPT];
  float  areaA[RPT], bi[RPT], bu[RPT];
  int    bidx[RPT], idx[RPT];
  bool   ok[RPT];
#pragma unroll
  for (int r = 0; r < RPT; ++r) {
    idx[r] = base + r * BLK;
    ok[r]  = idx[r] < n;
    float4 ar = ok[r] ? rois[idx[r]] : make_float4(0.f, 0.f, 0.f, 0.f);
    a[r] = ar;
    areaA[r] = (ar.z - ar.x + 1.0f) * (ar.w - ar.y + 1.0f);
    bi[r] = -1.0f;   // sentinel: first pair always wins (inter*1 > -1*uni)
    bu[r] =  1.0f;
    bidx[r] = 0;
  }

#pragma unroll 4
  for (int k = 0; k < mm; ++k) {
    const float4 g = sgt[k];
    const float  ab = sAreaB[k];
#pragma unroll
    for (int r = 0; r < RPT; ++r) {
      float ltx = fmaxf(a[r].x, g.x);
      float lty = fmaxf(a[r].y, g.y);
      float rbx = fminf(a[r].z, g.z);
      float rby = fminf(a[r].w, g.w);
      float w = fmaxf(rbx - ltx + 1.0f, 0.0f);
      float h = fmaxf(rby - lty + 1.0f, 0.0f);
      float inter = w * h;
      float uni = areaA[r] + ab - inter;
      // iou_k > iou_best  <=>  inter*bu > bi*uni  (unions positive);
      // strict '>' keeps earliest index on ties (matches jnp.argmax)
      bool better = inter * bu[r] > bi[r] * uni;
      bi[r]   = better ? inter : bi[r];
      bu[r]   = better ? uni   : bu[r];
      bidx[r] = better ? k     : bidx[r];
    }
  }

#pragma unroll
  for (int r = 0; r < RPT; ++r) {
    if (!ok[r]) continue;
    float mo = bi[r] / bu[r];       // single divide per roi
    maxiou[idx[r]] = mo;
    argm[idx[r]]   = bidx[r];
    float p = jax_priority((unsigned)idx[r], (unsigned)n);
    prio[idx[r]] = p;
    int bkt = bucket_of(p);
    if (mo > 0.5f)      atomicAdd(&fgHist[bkt], 1);
    else if (mo < 0.5f) atomicAdd(&bgHist[bkt], 1);
  }
}

// ---------------- kernel 2: radix-select thresholds ------------------
// meta: [0]=fgTotal [1]=bgTotal [2]=fgThresh [3]=bgThresh [4]=fgCandCnt
//       [5]=bgCandCnt [6]=numFg
__global__ void k_thresh(const int* __restrict__ fgHist,
                         const int* __restrict__ bgHist,
                         int* __restrict__ meta)
{
  int t = threadIdx.x;
  if (t >= 2) return;
  const int* h = (t == 0) ? fgHist : bgHist;
  const int  k = (t == 0) ? MAX_POS_C : TOTAL_SAMPLES_C;
  int total = 0;
  for (int b = 0; b < NBUCKET; ++b) total += h[b];
  int cum = 0, thresh = 0;
  for (int b = NBUCKET - 1; b >= 0; --b) {
    cum += h[b];
    if (cum >= k) { thresh = b; break; }
  }
  meta[t] = total;
  meta[2 + t] = thresh;
  if (t == 0) meta[6] = (total < MAX_POS_C) ? total : MAX_POS_C;
}

// ---------------- kernel 3: compact top-bucket candidates ------------
__global__ void k_collect(const float* __restrict__ maxiou,
                          const float* __restrict__ prio, int n,
                          int* __restrict__ meta,
                          int* __restrict__ fgCandIdx, float* __restrict__ fgCandPri,
                          int* __restrict__ bgCandIdx, float* __restrict__ bgCandPri)
{
  int i = blockIdx.x * blockDim.x + threadIdx.x;
  if (i >= n) return;
  float mo = maxiou[i];
  float p  = prio[i];
  int bkt  = bucket_of(p);
  if (mo > 0.5f) {
    if (bkt >= meta[2]) {
      int pos = atomicAdd(&meta[4], 1);
      if (pos < CAND_CAP) { fgCandIdx[pos] = i; fgCandPri[pos] = p; }
    }
  } else if (mo < 0.5f) {
    if (bkt >= meta[3]) {
      int pos = atomicAdd(&meta[5], 1);
      if (pos < CAND_CAP) { bgCandIdx[pos] = i; bgCandPri[pos] = p; }
    }
  }
}

// ---------------- kernel 4: exact rank of surviving candidates -------
// top_k ordering: descending value, ties broken by lower original index
__global__ __launch_bounds__(512) void k_rank(
    const int* __restrict__ meta,
    const int* __restrict__ fgCandIdx, const float* __restrict__ fgCandPri,
    const int* __restrict__ bgCandIdx, const float* __restrict__ bgCandPri,
    int* __restrict__ fgSel, int* __restrict__ bgSel)
{
  int t = threadIdx.x;
  int Lf = meta[4]; Lf = Lf < CAND_CAP ? Lf : CAND_CAP;
  int Lb = meta[5]; Lb = Lb < CAND_CAP ? Lb : CAND_CAP;

  for (int j = t; j < Lf; j += 512) {
    float pj = fgCandPri[j];
    int   ij = fgCandIdx[j];
    int rank = 0;
    for (int l = 0; l < Lf; ++l) {
      float pl = fgCandPri[l];
      rank += (pl > pj) || (pl == pj && fgCandIdx[l] < ij);
    }
    if (rank < MAX_POS_C) fgSel[rank] = ij;
  }
  for (int j = t; j < Lb; j += 512) {
    float pj = bgCandPri[j];
    int   ij = bgCandIdx[j];
    int rank = 0;
    for (int l = 0; l < Lb; ++l) {
      float pl = bgCandPri[l];
      rank += (pl > pj) || (pl == pj && bgCandIdx[l] < ij);
    }
    if (rank < TOTAL_SAMPLES_C) bgSel[rank] = ij;
  }
}

// ---------------- kernel 5: background fill of outputs ---------------
__global__ void k_fill(float* __restrict__ out, int total) {
  int i = blockIdx.x * blockDim.x + threadIdx.x;
  if (i < total) out[i] = (i >= OUTW_OFF) ? 1.0f : 0.0f;
}

// ---------------- kernel 6: per-slot outputs -------------------------
__global__ void k_out(const float4* __restrict__ rois, const float4* __restrict__ gt,
                      const int* __restrict__ gtl, const int* __restrict__ argm,
                      const int* __restrict__ meta,
                      const int* __restrict__ fgSel, const int* __restrict__ bgSel,
                      int n, int m, float* __restrict__ out)
{
  int slot = blockIdx.x * blockDim.x + threadIdx.x;
  if (slot >= TOTAL_SAMPLES_C) return;
  int numFg = meta[6];
  bool is_fg = slot < numFg;
  int keep;
  if (is_fg) {
    int s = slot < MAX_POS_C ? slot : MAX_POS_C - 1;
    keep = fgSel[s];
  } else {
    int bs = slot - numFg;
    bs = bs < 0 ? 0 : (bs > TOTAL_SAMPLES_C - 1 ? TOTAL_SAMPLES_C - 1 : bs);
    keep = bgSel[bs];
  }
  keep = keep < 0 ? 0 : (keep >= n ? n - 1 : keep);

  float4 roi = rois[keep];
  int g = argm[keep];
  g = g < 0 ? 0 : (g >= m ? m - 1 : g);
  int label = gtl[g];

  out[ROIS_OFF + slot * 4 + 0] = roi.x;
  out[ROIS_OFF + slot * 4 + 1] = roi.y;
  out[ROIS_OFF + slot * 4 + 2] = roi.z;
  out[ROIS_OFF + slot * 4 + 3] = roi.w;
  out[LBL_OFF + slot] = is_fg ? (float)label : 0.0f;

  if (is_fg) {
    float4 gb = gt[g];
    float rw  = fmaxf(roi.z - roi.x + 1.0f, 1e-6f);
    float rh  = fmaxf(roi.w - roi.y + 1.0f, 1e-6f);
    float rcx = roi.x + 0.5f * rw;
    float rcy = roi.y + 0.5f * rh;
    float gw  = fmaxf(gb.z - gb.x + 1.0f, 1e-6f);
    float gh  = fmaxf(gb.w - gb.y + 1.0f, 1e-6f);
    float gcx = gb.x + 0.5f * gw;
    float gcy = gb.y + 0.5f * gh;
    float t0 = (gcx - rcx) / rw;
    float t1 = (gcy - rcy) / rh;
    float t2 = logf(gw / rw);
    float t3 = logf(gh / rh);
    int lb = label;
    lb = lb < 0 ? 0 : (lb >= NUM_CLASSES_C ? NUM_CLASSES_C - 1 : lb);
    int tb = TGT_OFF + slot * (NUM_CLASSES_C * 4) + lb * 4;
    out[tb + 0] = t0; out[tb + 1] = t1; out[tb + 2] = t2; out[tb + 3] = t3;
    int ib = INS_OFF + slot * (NUM_CLASSES_C * 4) + lb * 4;
    out[ib + 0] = 1.0f; out[ib + 1] = 1.0f; out[ib + 2] = 1.0f; out[ib + 3] = 1.0f;
  }
}

// ---------------- launch -------------------------------------------
extern "C" void kernel_launch(void* const* d_in, const int* in_sizes, int n_in,
                              void* d_out, int out_size, void* d_ws, size_t ws_size,
                              hipStream_t stream) {
  const float4* rois = (const float4*)d_in[0];
  const float4* gt   = (const float4*)d_in[1];
  const int*    gtl  = (const int*)d_in[2];
  const int N = in_sizes[0] / 4;
  const int M = in_sizes[1] / 4;

  // carve workspace (256B aligned slices)
  char* w = (char*)d_ws;
  auto carve = [&](size_t bytes) {
    void* p = (void*)w;
    w += (bytes + 255) & ~(size_t)255;
    return p;
  };
  float* maxiou    = (float*)carve((size_t)N * sizeof(float));
  int*   argm      = (int*)  carve((size_t)N * sizeof(int));
  float* prio      = (float*)carve((size_t)N * sizeof(float));
  char*  zbase     = w;                       // zeroed control region starts here
  int*   fgHist    = (int*)  carve(NBUCKET * sizeof(int));
  int*   bgHist    = (int*)  carve(NBUCKET * sizeof(int));
  int*   meta      = (int*)  carve(64 * sizeof(int));
  int*   fgCandIdx = (int*)  carve(CAND_CAP * sizeof(int));
  float* fgCandPri = (float*)carve(CAND_CAP * sizeof(float));
  int*   bgCandIdx = (int*)  carve(CAND_CAP * sizeof(int));
  float* bgCandPri = (float*)carve(CAND_CAP * sizeof(float));
  int*   fgSel     = (int*)  carve(MAX_POS_C * sizeof(int));
  int*   bgSel     = (int*)  carve(TOTAL_SAMPLES_C * sizeof(int));
  int zero_ints = (int)((w - zbase) / 4);

  // 0) zero histograms / meta / candidate lists / selections
  k_init<<<(zero_ints + 255) / 256, 256, 0, stream>>>((int*)zbase, zero_ints);

  // 1) IoU max/argmax + threefry priority + radix histogram (dominant)
  int iouBlocks = (N + BLK * RPT - 1) / (BLK * RPT);
  k_iou<<<iouBlocks, BLK, 0, stream>>>(rois, gt, N, M, maxiou, argm, prio,
                                       fgHist, bgHist);

  // 2) descending-scan thresholds for top-128 fg / top-512 bg
  k_thresh<<<1, 64, 0, stream>>>(fgHist, bgHist, meta);

  // 3) compact candidates in winning buckets
  k_collect<<<(N + 255) / 256, 256, 0, stream>>>(maxiou, prio, N, meta,
                                                 fgCandIdx, fgCandPri,
                                                 bgCandIdx, bgCandPri);

  // 4) exact top-k ordering among candidates
  k_rank<<<1, 512, 0, stream>>>(meta, fgCandIdx, fgCandPri,
                                bgCandIdx, bgCandPri, fgSel, bgSel);

  // 5) fill outputs: zeros for targets/inside, ones for outside weights
  int total = out_size < OUT_TOTAL ? out_size : OUT_TOTAL;
  k_fill<<<(total + 255) / 256, 256, 0, stream>>>((float*)d_out, total);

  // 6) per-slot rois / labels / encoded targets / inside weights
  k_out<<<(TOTAL_SAMPLES_C + 255) / 256, 256, 0, stream>>>(
      rois, gt, gtl, argm, meta, fgSel, bgSel, N, M, (float*)d_out);
}